// GraphConvolution_82403242541780
// MI455X (gfx1250) — compile-verified
//
#include <hip/hip_runtime.h>

// ---------------------------------------------------------------------------
// GraphConvolution for MI455X (gfx1250, wave32)
//   support = feat @ W   (WMMA f32 16x16x4; async-to-LDS staged A; 5 M-tiles
//                         per wave reuse each B fragment)
//   out     = bias + scatter_add(edge_val * support[edge_col]) over edge_row
//             (32 edges/wave batched; v_readlane broadcast of edge scalars)
// ---------------------------------------------------------------------------

typedef __attribute__((ext_vector_type(2))) float v2f;
typedef __attribute__((ext_vector_type(8))) float v8f;

#define N_NODES  100000
#define N_EDGES  3200000
#define IN_DIM   256
#define OUT_DIM  128

#define ROWS_PER_BLOCK 80      // 5 M-tiles of 16; 100000 = 80 * 1250 exactly
#define M_TILES        5

// LDS row stride for the 80x256 A stripe: 256 + 4 pad floats.
// A-fragment read: lane l does a b64 load at float index (l&15)*260 + kb,
// kb = k + 2*(l>>4).  Dword bank = 4*(l&15) + kb mod 64 -> all 64 banks
// touched exactly once: conflict-free.  Row stride 1040 B is 8B-aligned,
// legal for both ds_load_b64 and the b64 async-to-LDS stores.
#define LDS_STRIDE 260

// --------------------------- bias broadcast --------------------------------
__global__ __launch_bounds__(256) void GC_bias_init(
    float* __restrict__ out, const float* __restrict__ bias) {
  // one float4 per thread: 100000*128/4 = 3,200,000 float4s
  int idx = blockIdx.x * blockDim.x + threadIdx.x;
  int c4 = (idx & 31) * 4;
  float4 b = *(const float4*)(bias + c4);
  *(float4*)(out + (size_t)idx * 4) = b;
}

// --------------------------- GEMM (WMMA f32) -------------------------------
__global__ __launch_bounds__(256) void GC_gemm_wmma(
    const float* __restrict__ feat, const float* __restrict__ weight,
    float* __restrict__ support) {
  extern __shared__ float ldsA[];            // 80 * 260 * 4 = 83,200 B

  const int m0 = blockIdx.x * ROWS_PER_BLOCK;
  const int t  = threadIdx.x;

  // ---- Phase 1: async-copy the 80x256 feat stripe into LDS ----------------
  // 80*128 = 10240 b64 transfers; 40 per thread.  Wave-contiguous global
  // addresses (256B/wave per instruction).  Tracked by ASYNCcnt.
  const uint32_t lds0 = (uint32_t)(uintptr_t)(void*)&ldsA[0];  // LDS byte offset
#pragma unroll 4
  for (int i = 0; i < 40; ++i) {
    int idx  = i * 256 + t;                  // b64 index in stripe
    int row  = idx >> 7;                     // / 128 b64s per row
    int col2 = idx & 127;                    // b64 column
    uint64_t g = (uint64_t)(uintptr_t)(feat + (size_t)(m0 + row) * IN_DIM + col2 * 2);
    uint32_t l = lds0 + (uint32_t)((row * LDS_STRIDE + col2 * 2) * 4);
    asm volatile("global_load_async_to_lds_b64 %0, %1, off"
                 :: "v"(l), "v"(g) : "memory");
  }
  asm volatile("s_wait_asynccnt 0x0" ::: "memory");
  __syncthreads();

  // ---- Phase 2: each wave computes an 80x16 column slab (5 WMMA tiles) ----
  const int lane = t & 31;
  const int wv   = t >> 5;                   // 0..7 -> N tile
  const int n0   = wv * 16;
  const int mm   = lane & 15;                // A row / B col within tile
  const int kk   = (lane >> 4) << 1;         // 0 or 2: K sub-offset per half

  v8f acc[M_TILES];
#pragma unroll
  for (int mt = 0; mt < M_TILES; ++mt) acc[mt] = v8f{};

  const float* bcol = weight + (size_t)kk * OUT_DIM + (n0 + mm);
  const float* arow = &ldsA[mm * LDS_STRIDE + kk];

#pragma unroll 2
  for (int k = 0; k < IN_DIM; k += 4) {
    // B fragment: 4x16 f32, shared by all 5 M tiles (2 global b32 loads)
    v2f b;
    b[0] = bcol[(size_t)k * OUT_DIM];
    b[1] = bcol[(size_t)(k + 1) * OUT_DIM];
    // A fragments: 5 conflict-free ds_load_b64
    v2f a[M_TILES];
#pragma unroll
    for (int mt = 0; mt < M_TILES; ++mt)
      a[mt] = *(const v2f*)(arow + mt * 16 * LDS_STRIDE + k);
#pragma unroll
    for (int mt = 0; mt < M_TILES; ++mt)
      acc[mt] = __builtin_amdgcn_wmma_f32_16x16x4_f32(
          /*neg_a=*/false, a[mt], /*neg_b=*/false, b,
          /*c_mod=*/(short)0, acc[mt], /*reuse_a=*/false, /*reuse_b=*/false);
  }

  // D layout: VGPR v holds row (v + 8*(lane>>4)), col = lane&15.
  const int colg  = n0 + mm;
  const int rhalf = (lane >> 4) << 3;
#pragma unroll
  for (int mt = 0; mt < M_TILES; ++mt) {
    const int rbase = m0 + mt * 16 + rhalf;
#pragma unroll
    for (int v = 0; v < 8; ++v)
      support[(size_t)(rbase + v) * OUT_DIM + colg] = acc[mt][v];
  }
}

// --------------------------- SpMM scatter-add ------------------------------
// 32 edges per wave: each lane loads its own edge's (row,col,val) coalesced,
// then v_readlane broadcasts each edge's scalars (wave-uniform -> SGPR
// addressing).  Per edge: 1 coalesced b128 gather + 4 global_atomic_add_f32.
// N_EDGES = 100000 * 32, so 100000 waves cover all edges with no tail.
__global__ __launch_bounds__(256) void GC_spmm_atomic(
    const int* __restrict__ edge_row, const int* __restrict__ edge_col,
    const float* __restrict__ edge_val, const float* __restrict__ support,
    float* __restrict__ out) {
  const int tid  = blockIdx.x * blockDim.x + threadIdx.x;
  const int lane = tid & 31;
  const int wave = tid >> 5;                  // [0, 100000)

  const int e = wave * 32 + lane;             // this lane's edge of the batch
  const int   rv = edge_row[e];               // coalesced b32
  const int   cv = edge_col[e];               // coalesced b32
  const float vv = edge_val[e];               // coalesced b32
  const int   vvi = __float_as_int(vv);

  const int laneOff = lane * 4;               // 4 output floats per lane

#pragma unroll
  for (int j = 0; j < 32; ++j) {
    const int   r = __builtin_amdgcn_readlane(rv, j);   // uniform (SGPR)
    const int   c = __builtin_amdgcn_readlane(cv, j);   // uniform (SGPR)
    const float v = __int_as_float(__builtin_amdgcn_readlane(vvi, j));
    // coalesced 512B gather of support row c (L2-resident: 51.2MB < 192MB L2)
    const float4 g = *(const float4*)(support + (size_t)c * OUT_DIM + laneOff);
    float* o = out + (size_t)r * OUT_DIM + laneOff;
    // non-returning global_atomic_add_f32 x4 (STOREcnt), L2-resident output
    unsafeAtomicAdd(o + 0, v * g.x);
    unsafeAtomicAdd(o + 1, v * g.y);
    unsafeAtomicAdd(o + 2, v * g.z);
    unsafeAtomicAdd(o + 3, v * g.w);
  }
}

// --------------------------- launcher --------------------------------------
extern "C" void kernel_launch(void* const* d_in, const int* in_sizes, int n_in,
                              void* d_out, int out_size, void* d_ws, size_t ws_size,
                              hipStream_t stream) {
  const float* feat     = (const float*)d_in[0];
  const int*   edge_row = (const int*)  d_in[1];
  const int*   edge_col = (const int*)  d_in[2];
  const float* edge_val = (const float*)d_in[3];
  const float* weight   = (const float*)d_in[4];
  const float* bias     = (const float*)d_in[5];
  float*       out      = (float*)d_out;
  float*       support  = (float*)d_ws;   // 100000*128*4 = 51.2 MB scratch

  // out = bias (also resets the accumulator every graph replay)
  GC_bias_init<<<(N_NODES * OUT_DIM / 4) / 256, 256, 0, stream>>>(out, bias);

  // support = feat @ W : 1250 blocks x (80 rows, 8 waves x 16-col slabs)
  const size_t ldsBytes = (size_t)ROWS_PER_BLOCK * LDS_STRIDE * sizeof(float);
  GC_gemm_wmma<<<N_NODES / ROWS_PER_BLOCK, 256, ldsBytes, stream>>>(
      feat, weight, support);

  // out += scatter_add(edge_val * support[edge_col])
  // 3.2M edges / 32-per-wave = 100000 waves = 12500 blocks of 8 waves
  GC_spmm_atomic<<<N_EDGES / 32 / 8, 256, 0, stream>>>(edge_row, edge_col,
                                                       edge_val, support, out);
}